// GINBlock_8126078124213
// MI455X (gfx1250) — compile-verified
//
#include <hip/hip_runtime.h>

typedef float v2f __attribute__((ext_vector_type(2)));
typedef float v8f __attribute__((ext_vector_type(8)));

#define CH 128

// ---------- monotonic float<->uint encoding for atomic max ----------
__device__ __forceinline__ unsigned int enc_f32(float f) {
    unsigned int u = __float_as_uint(f);
    return (u & 0x80000000u) ? ~u : (u | 0x80000000u);
}
__device__ __forceinline__ float dec_f32(unsigned int e) {
    unsigned int u = (e & 0x80000000u) ? (e & 0x7FFFFFFFu) : ~e;
    return __uint_as_float(u);
}

// ---------- init: clear AGG (sentinel 0 = empty) and LN stats ----------
__global__ void init_kernel(unsigned int* __restrict__ agg, double* __restrict__ stats,
                            long long total4) {
    long long i = (long long)blockIdx.x * blockDim.x + threadIdx.x;
    if (i < total4) {
        uint4 z = make_uint4(0u, 0u, 0u, 0u);
        ((uint4*)agg)[i] = z;
    }
    if (i == 0) { stats[0] = 0.0; stats[1] = 0.0; }
}

// ---------- per-edge gather + atomic segment max (32 threads / edge) ----------
__global__ void edge_kernel(const float* __restrict__ x,
                            const int* __restrict__ src,
                            const int* __restrict__ dst,
                            unsigned int* __restrict__ agg, int E) {
    long long gid = (long long)blockIdx.x * blockDim.x + threadIdx.x;
    int e  = (int)(gid >> 5);
    int c4 = ((int)gid & 31) * 4;
    if (e >= E) return;
    int s = src[e];
    int d = dst[e];
    const float4 v = *(const float4*)(x + (size_t)s * CH + c4);
    unsigned int* base = agg + (size_t)d * CH + c4;
    atomicMax(base + 0, enc_f32(v.x));
    atomicMax(base + 1, enc_f32(v.y));
    atomicMax(base + 2, enc_f32(v.z));
    atomicMax(base + 3, enc_f32(v.w));
}

// ---------- h = (1+eps)*x + agg (empty segment -> 0) ----------
__global__ void fuse_kernel(const float* __restrict__ x,
                            const unsigned int* __restrict__ agg,
                            const float* __restrict__ epsPtr,
                            float* __restrict__ h, long long total) {
    long long i = (long long)blockIdx.x * blockDim.x + threadIdx.x;
    if (i >= total) return;
    float eps = epsPtr[0];
    unsigned int e = agg[i];
    float a = (e == 0u) ? 0.0f : dec_f32(e);
    h[i] = (1.0f + eps) * x[i] + a;
}

// ---------- WMMA fp32 GEMM: out[N,128] = A[N,128] @ W^T + b ----------
// One wave computes a 16-row strip across all 8 column tiles.
// A fragment (16x4 f32): lane lo holds rows, hi selects K pair -> contiguous float2.
// B fragment (4x16 f32): element (K, N) = W[N][K]        -> contiguous float2.
__global__ __launch_bounds__(256) void gemm_kernel(const float* __restrict__ A,
                                                   const float* __restrict__ W,
                                                   const float* __restrict__ bias,
                                                   float* __restrict__ out,
                                                   double* __restrict__ stats,
                                                   int doStats, int nRowTiles, int N) {
    int wave = threadIdx.x >> 5;
    int lane = threadIdx.x & 31;
    int tile = blockIdx.x * (blockDim.x >> 5) + wave;
    if (tile >= nRowTiles) return;          // wave-uniform branch

    int row0 = tile * 16;
    int lo = lane & 15;
    int hi = lane >> 4;

    v8f acc[8];
#pragma unroll
    for (int t = 0; t < 8; ++t) {
        float bv = bias[t * 16 + lo];
#pragma unroll
        for (int v = 0; v < 8; ++v) acc[t][v] = bv;
    }

    int rowA = row0 + lo;
    if (rowA >= N) rowA = N - 1;            // clamp so every lane loads valid memory
    const float* arow = A + (size_t)rowA * CH + 2 * hi;

#pragma unroll 4
    for (int k = 0; k < CH; k += 4) {
        v2f a = *(const v2f*)(arow + k);
#pragma unroll
        for (int t = 0; t < 8; ++t) {
            v2f bfrag = *(const v2f*)(W + (size_t)(t * 16 + lo) * CH + k + 2 * hi);
            acc[t] = __builtin_amdgcn_wmma_f32_16x16x4_f32(
                /*neg_a=*/false, a, /*neg_b=*/false, bfrag,
                /*c_mod=*/(short)0, acc[t], /*reuse_a=*/false, /*reuse_b=*/false);
        }
    }

    float s = 0.0f, sq = 0.0f;
#pragma unroll
    for (int t = 0; t < 8; ++t) {
#pragma unroll
        for (int v = 0; v < 8; ++v) {
            int row = row0 + v + 8 * hi;    // D layout: VGPR v -> M = v + 8*hi
            float val = acc[t][v];
            if (row < N) out[(size_t)row * CH + t * 16 + lo] = val;
            s += val; sq += val * val;
        }
    }

    if (doStats) {
#pragma unroll
        for (int off = 16; off > 0; off >>= 1) {
            s  += __shfl_down(s,  off, 32);
            sq += __shfl_down(sq, off, 32);
        }
        if (lane == 0) {
            atomicAdd(&stats[0], (double)s);
            atomicAdd(&stats[1], (double)sq);
        }
    }
}

// ---------- LN stats: mean and 1/(std + eps) ----------
__global__ void stats_kernel(const double* __restrict__ stats,
                             float* __restrict__ meanInv, long long total) {
    double mean = stats[0] / (double)total;
    double var  = stats[1] / (double)total - mean * mean;
    if (var < 0.0) var = 0.0;
    float stdv = (float)sqrt(var);
    meanInv[0] = (float)mean;
    meanInv[1] = 1.0f / (stdv + 1e-5f);
}

// ---------- graph LayerNorm + PReLU ----------
__global__ void ln_prelu_kernel(const float* __restrict__ z,
                                const float* __restrict__ lnw,
                                const float* __restrict__ lnb,
                                const float* __restrict__ aPtr,
                                const float* __restrict__ meanInv,
                                float* __restrict__ h, long long total) {
    long long i = (long long)blockIdx.x * blockDim.x + threadIdx.x;
    if (i >= total) return;
    int c = (int)(i & (CH - 1));
    float m = meanInv[0], inv = meanInv[1];
    float y = (z[i] - m) * inv * lnw[c] + lnb[c];
    float a = aPtr[0];
    h[i] = (y >= 0.0f) ? y : a * y;
}

extern "C" void kernel_launch(void* const* d_in, const int* in_sizes, int n_in,
                              void* d_out, int out_size, void* d_ws, size_t ws_size,
                              hipStream_t stream) {
    const float* x    = (const float*)d_in[0];
    const int*   ei   = (const int*)d_in[1];
    const float* eps1 = (const float*)d_in[2];
    const float* W1   = (const float*)d_in[3];
    const float* b1   = (const float*)d_in[4];
    const float* lnw  = (const float*)d_in[5];
    const float* lnb  = (const float*)d_in[6];
    const float* pa   = (const float*)d_in[7];
    const float* eps2 = (const float*)d_in[8];
    const float* W2   = (const float*)d_in[9];
    const float* b2   = (const float*)d_in[10];

    const int N = in_sizes[0] / CH;
    const int E = in_sizes[1] / 2;
    const int* src = ei;
    const int* dst = ei + E;

    const long long nc = (long long)N * CH;

    // workspace layout
    unsigned int* AGG  = (unsigned int*)d_ws;
    float* H1    = (float*)((char*)d_ws + (size_t)nc * 4);          // layer1 pre-GEMM / layer2 pre-GEMM
    float* Z1    = (float*)((char*)d_ws + (size_t)nc * 8);          // layer1 GEMM out
    float* H2    = (float*)((char*)d_ws + (size_t)nc * 12);         // post LN+PReLU
    double* stats = (double*)((char*)d_ws + (size_t)nc * 16);
    float* meanInv = (float*)(stats + 2);

    const int elemBlocks = (int)((nc + 255) / 256);
    const int initBlocks = (int)((nc / 4 + 255) / 256);
    const int edgeBlocks = (int)(((long long)E * 32 + 255) / 256);
    const int nRowTiles  = (N + 15) / 16;
    const int gemmBlocks = (nRowTiles + 7) / 8;

    // ---- layer 1: GIN(max) -> linear ----
    init_kernel<<<initBlocks, 256, 0, stream>>>(AGG, stats, nc / 4);
    edge_kernel<<<edgeBlocks, 256, 0, stream>>>(x, src, dst, AGG, E);
    fuse_kernel<<<elemBlocks, 256, 0, stream>>>(x, AGG, eps1, H1, nc);
    gemm_kernel<<<gemmBlocks, 256, 0, stream>>>(H1, W1, b1, Z1, stats, 1, nRowTiles, N);

    // ---- graph LayerNorm + PReLU ----
    stats_kernel<<<1, 1, 0, stream>>>(stats, meanInv, nc);
    ln_prelu_kernel<<<elemBlocks, 256, 0, stream>>>(Z1, lnw, lnb, pa, meanInv, H2, nc);

    // ---- layer 2: GIN(max) -> linear -> d_out ----
    init_kernel<<<initBlocks, 256, 0, stream>>>(AGG, stats, nc / 4);
    edge_kernel<<<edgeBlocks, 256, 0, stream>>>(H2, src, dst, AGG, E);
    fuse_kernel<<<elemBlocks, 256, 0, stream>>>(H2, AGG, eps2, H1, nc);
    gemm_kernel<<<gemmBlocks, 256, 0, stream>>>(H1, W2, b2, (float*)d_out, stats, 0, nRowTiles, N);
}